// TorchNeighborListRangeSeparatedMLMM_5214090297976
// MI455X (gfx1250) — compile-verified
//
#include <hip/hip_runtime.h>
#include <hip/hip_bf16.h>
#include <stdint.h>

#define MAX_ML 128
#define MAX_CUT 4

typedef float v2f __attribute__((ext_vector_type(2)));
typedef float v8f __attribute__((ext_vector_type(8)));
typedef unsigned int u32x4 __attribute__((ext_vector_type(4)));
typedef int i32x4 __attribute__((ext_vector_type(4)));
typedef int i32x8 __attribute__((ext_vector_type(8)));

#if defined(__gfx1250__) && __has_builtin(__builtin_amdgcn_tensor_load_to_lds)
#define HAVE_TDM 1
#else
#define HAVE_TDM 0
#endif

#if defined(__gfx1250__) && __has_builtin(__builtin_amdgcn_wmma_f32_16x16x4_f32)
#define HAVE_WMMA_F32X4 1
#else
#define HAVE_WMMA_F32X4 0
#endif

#if HAVE_TDM
// 1-D contiguous dword copy (global -> LDS) via Tensor Data Mover.
// D# group0: count=1, lds_addr, global_addr[56:0], type=2.
// D# group1: data_size=4B, tensor_dim0 = nd, tensor_dim1 = 1,
//            tile_dim0 = nd (<=65535), tile_dim1 = 1, dim0_stride = nd.
// 6-arg builtin on this toolchain: (g0, g1, g2, g3, gx, cpol).
__device__ __forceinline__ void tdm_load_1d_dwords(uint64_t gaddr, uint32_t lds_off, uint32_t nd) {
  u32x4 g0;
  g0[0] = 1u;                                                   // count=1, user mode
  g0[1] = lds_off;                                              // lds_addr (bytes)
  g0[2] = (uint32_t)(gaddr & 0xFFFFFFFFull);                    // global_addr[31:0]
  g0[3] = ((uint32_t)(gaddr >> 32) & 0x01FFFFFFu) | (2u << 30); // addr[56:32] | type=2
  i32x8 g1;
  g1[0] = 0x00020000;                                   // data_size = 4 bytes
  g1[1] = (int)((nd & 0xFFFFu) << 16);                  // tensor_dim0[15:0]
  g1[2] = (int)(((nd >> 16) & 0xFFFFu) | (1u << 16));   // tensor_dim0[31:16] | tensor_dim1=1
  g1[3] = (int)((nd & 0xFFFFu) << 16);                  // tile_dim0 = nd
  g1[4] = 1;                                            // tile_dim1 = 1, tile_dim2 = 0
  g1[5] = (int)nd;                                      // tensor_dim0_stride[31:0]
  g1[6] = 0;
  g1[7] = 0;
  i32x4 z4; z4[0] = z4[1] = z4[2] = z4[3] = 0;          // groups 2/3 unused (<=2D)
  i32x8 z8; z8[0]=z8[1]=z8[2]=z8[3]=z8[4]=z8[5]=z8[6]=z8[7]=0;
  __builtin_amdgcn_tensor_load_to_lds(g0, g1, z4, z4, z8, 0);
}
#endif

__device__ __forceinline__ void nt_store(float* p, float v) {
  __builtin_nontemporal_store(v, p);
}

// Output layout (flat concat, element counts), P = (S+1)*n_ml*n_mm:
//   idx_i        [0      , 2P)
//   idx_j        [2P     , 4P)
//   offsets_cart [4P     , 10P)   row-major (2P,3)
//   d2_full      [10P    , 12P)
//   masks        [12P    , 12P + nc*2P)  row-major (nc, 2P), 1.0/0.0
__global__ __launch_bounds__(256)
void nlist_wmma_kernel(const float* __restrict__ pos,
                       const float* __restrict__ cell,
                       const float* __restrict__ cutoffs,
                       const int*   __restrict__ shifts,
                       const int*   __restrict__ ml_idx,
                       const int*   __restrict__ mm_idx,
                       float*       __restrict__ out,
                       int n_ml, int n_mm, int S, int nc)
{
  __shared__ float s_mlpos[MAX_ML * 3];
  __shared__ int   s_mlidx[MAX_ML];
  __shared__ float s_mmpos[16 * 3];
  __shared__ int   s_mmidx[16];
  __shared__ int   s_ish[64];     // TDM dest: shifts (S*3 ints, padded)
  __shared__ float s_cell[12];    // TDM dest: cell (9 floats, padded)

  const int t  = threadIdx.x;
  const int j0 = (int)blockIdx.x * 16;
  const int s  = (int)blockIdx.y;           // image index 0..S  (0 = zero shift)
  if (j0 >= n_mm) return;                   // uniform per block

  // ---- Stage tiny inputs into LDS --------------------------------------
#if HAVE_TDM
  if (t < 32) {  // wave 0 only; TDM ignores EXEC, issues once per wave
    tdm_load_1d_dwords((uint64_t)(uintptr_t)shifts,
                       (uint32_t)(uintptr_t)&s_ish[0],  (uint32_t)(S * 3));
    tdm_load_1d_dwords((uint64_t)(uintptr_t)cell,
                       (uint32_t)(uintptr_t)&s_cell[0], 9u);
    __builtin_amdgcn_s_wait_tensorcnt(0);
  }
#else
  if (t < S * 3 && t < 64) s_ish[t] = shifts[t];
  if (t >= 64 && t < 73)   s_cell[t - 64] = cell[t - 64];
#endif
  for (int r = t; r < n_ml && r < MAX_ML; r += 256) {
    int gi = ml_idx[r];
    s_mlidx[r] = gi;
    s_mlpos[r * 3 + 0] = pos[gi * 3 + 0];
    s_mlpos[r * 3 + 1] = pos[gi * 3 + 1];
    s_mlpos[r * 3 + 2] = pos[gi * 3 + 2];
  }
  if (t < 16) {
    int jj = j0 + t;
    int gj = mm_idx[(jj < n_mm) ? jj : 0];
    s_mmidx[t] = gj;
    s_mmpos[t * 3 + 0] = pos[gj * 3 + 0];
    s_mmpos[t * 3 + 1] = pos[gj * 3 + 1];
    s_mmpos[t * 3 + 2] = pos[gj * 3 + 2];
  }
  __syncthreads();

  // shift_vec[s] = shift_frac[s] @ cell  (row 0 is the zero shift)
  float fx = 0.f, fy = 0.f, fz = 0.f;
  if (s > 0) {
    fx = (float)s_ish[(s - 1) * 3 + 0];
    fy = (float)s_ish[(s - 1) * 3 + 1];
    fz = (float)s_ish[(s - 1) * 3 + 2];
  }
  const float svx = fx * s_cell[0] + fy * s_cell[3] + fz * s_cell[6];
  const float svy = fx * s_cell[1] + fy * s_cell[4] + fz * s_cell[7];
  const float svz = fx * s_cell[2] + fy * s_cell[5] + fz * s_cell[8];

  float cut2[MAX_CUT];
  for (int c = 0; c < nc && c < MAX_CUT; ++c) { float cv = cutoffs[c]; cut2[c] = cv * cv; }

  const int lane = t & 31;
  const int wv   = t >> 5;
  const int half = lane >> 4;       // 0: lanes 0-15, 1: lanes 16-31
  const int ln   = lane & 15;
  const int k0   = half * 2;        // K pair handled by this half-wave

  const long P0 = (long)n_ml * (long)n_mm;
  const long P  = (long)(S + 1) * P0;
  const uint32_t Pu  = (uint32_t)P;
  const uint32_t dq  = (uint32_t)n_mm;        // q advance per D-row
  const uint32_t dq3 = 3u * (uint32_t)n_mm;

  // Per-section SGPR base pointers; all in-section offsets fit in u32,
  // so stores lower to the SADDR+u32-voffset (GVS) form.
  float* const sec_ii_lo = out;               // idx_i[q]
  float* const sec_ii_hi = out + P;           // idx_i[P+q]
  float* const sec_ij_lo = out + 2 * P;       // idx_j[q]
  float* const sec_ij_hi = out + 3 * P;       // idx_j[P+q]
  float* const sec_of_lo = out + 4 * P;       // offsets_cart[q]   (+3q)
  float* const sec_of_hi = out + 7 * P;       // offsets_cart[P+q] (+3q)
  float* const sec_d2_lo = out + 10 * P;      // d2_full[q]
  float* const sec_d2_hi = out + 11 * P;      // d2_full[P+q]
  float* const sec_mk    = out + 12 * P;      // masks base

  // B operand (shared by all i-tiles of this wave): col n = ln
  const float bx = s_mmpos[ln * 3 + 0];
  const float by = s_mmpos[ln * 3 + 1];
  const float bz = s_mmpos[ln * 3 + 2];
  const float bcomp[4] = { -2.f * bx, -2.f * by, -2.f * bz, bx * bx + by * by + bz * bz };
  v2f B; B[0] = bcomp[k0]; B[1] = bcomp[k0 + 1];
  const float jval = (float)s_mmidx[ln];
  const int   jcol = j0 + ln;

  // 8 waves cover i-tiles 0..7 (n_ml = 128); stride for generality
  for (int i0 = wv * 16; i0 + 16 <= n_ml; i0 += 16 * 8) {
    // A operand: row m = ln, shifted ML position, 4th component = 1
    const float ax = s_mlpos[(i0 + ln) * 3 + 0] + svx;
    const float ay = s_mlpos[(i0 + ln) * 3 + 1] + svy;
    const float az = s_mlpos[(i0 + ln) * 3 + 2] + svz;
    const float acomp[4] = { ax, ay, az, 1.f };
    v2f A; A[0] = acomp[k0]; A[1] = acomp[k0 + 1];

    // C operand: |a_m|^2 for m = r + half*8
    v8f C;
#pragma unroll
    for (int r = 0; r < 8; ++r) {
      const int m = r + half * 8;
      const float cx = s_mlpos[(i0 + m) * 3 + 0] + svx;
      const float cy = s_mlpos[(i0 + m) * 3 + 1] + svy;
      const float cz = s_mlpos[(i0 + m) * 3 + 2] + svz;
      C[r] = cx * cx + cy * cy + cz * cz;
    }

    v8f D;
#if HAVE_WMMA_F32X4
    // d2[m][n] = |a_m|^2 - 2 a_m.b_n + |b_n|^2  via 16x16x4 f32 WMMA
    D = __builtin_amdgcn_wmma_f32_16x16x4_f32(false, A, false, B,
                                              (short)0, C, false, false);
#else
#pragma unroll
    for (int r = 0; r < 8; ++r) {
      const int m = r + half * 8;
      const float dx = (s_mlpos[(i0 + m) * 3 + 0] + svx) - bx;
      const float dy = (s_mlpos[(i0 + m) * 3 + 1] + svy) - by;
      const float dz = (s_mlpos[(i0 + m) * 3 + 2] + svz) - bz;
      D[r] = dx * dx + dy * dy + dz * dz;
    }
#endif

    // ---- Streaming NT stores: 16 output floats per pair ----------------
    // This lane owns D rows m = half*8 + r  ->  q advances by n_mm per row.
    const int mbase = i0 + half * 8;
    uint32_t q  = (uint32_t)((long)s * P0 + (long)mbase * (long)n_mm + (long)jcol);
    uint32_t q3 = 3u * q;

#pragma unroll
    for (int r = 0; r < 8; ++r) {
      const float ival = (float)s_mlidx[mbase + r];
      const float d2v  = D[r];

      nt_store(sec_ii_lo + q, ival);
      nt_store(sec_ii_hi + q, jval);
      nt_store(sec_ij_lo + q, jval);
      nt_store(sec_ij_hi + q, ival);

      nt_store(sec_of_lo + q3 + 0, -svx);
      nt_store(sec_of_lo + q3 + 1, -svy);
      nt_store(sec_of_lo + q3 + 2, -svz);
      nt_store(sec_of_hi + q3 + 0,  svx);
      nt_store(sec_of_hi + q3 + 1,  svy);
      nt_store(sec_of_hi + q3 + 2,  svz);

      nt_store(sec_d2_lo + q, d2v);
      nt_store(sec_d2_hi + q, d2v);

      if (nc == 2) {             // fast path: both cutoffs unrolled
        const float m0 = (d2v < cut2[0]) ? 1.0f : 0.0f;
        const float m1 = (d2v < cut2[1]) ? 1.0f : 0.0f;
        nt_store(sec_mk + q,           m0);
        nt_store(sec_mk + Pu + q,      m0);
        nt_store(sec_mk + 2u * Pu + q, m1);
        nt_store(sec_mk + 3u * Pu + q, m1);
      } else {
        for (int c = 0; c < nc && c < MAX_CUT; ++c) {
          const float mv = (d2v < cut2[c]) ? 1.0f : 0.0f;
          nt_store(sec_mk + (uint32_t)(2 * c) * Pu + q,      mv);
          nt_store(sec_mk + (uint32_t)(2 * c + 1) * Pu + q,  mv);
        }
      }

      q  += dq;
      q3 += dq3;
    }
  }
}

extern "C" void kernel_launch(void* const* d_in, const int* in_sizes, int n_in,
                              void* d_out, int out_size, void* d_ws, size_t ws_size,
                              hipStream_t stream) {
  (void)n_in; (void)out_size; (void)d_ws; (void)ws_size;
  const float* pos     = (const float*)d_in[0];
  const float* cell    = (const float*)d_in[1];
  const float* cutoffs = (const float*)d_in[2];
  const int*   shifts  = (const int*)d_in[3];
  const int*   ml_idx  = (const int*)d_in[4];
  const int*   mm_idx  = (const int*)d_in[5];

  const int n_ml = in_sizes[4];
  const int n_mm = in_sizes[5];
  const int S    = in_sizes[3] / 3;   // number of half-shifts (13)
  const int nc   = in_sizes[2];       // number of cutoffs (2)

  dim3 grid((unsigned)((n_mm + 15) / 16), (unsigned)(S + 1));
  nlist_wmma_kernel<<<grid, 256, 0, stream>>>(pos, cell, cutoffs, shifts,
                                              ml_idx, mm_idx, (float*)d_out,
                                              n_ml, n_mm, S, nc);
}